// PixelHopPlusPlus_54391465836771
// MI455X (gfx1250) — compile-verified
//
#include <hip/hip_runtime.h>

typedef float v2f __attribute__((ext_vector_type(2)));
typedef float v8f __attribute__((ext_vector_type(8)));

// D(16x16,f32) = A(16x4,f32) * B(4x16,f32) + C ; exact fp32 math.
__device__ __forceinline__ v8f wmma4(v2f a, v2f b, v8f c) {
  return __builtin_amdgcn_wmma_f32_16x16x4_f32(
      /*neg_a=*/false, a, /*neg_b=*/false, b,
      /*c_mod=*/(short)0, c, /*reuse_a=*/false, /*reuse_b=*/false);
}

// Fused PixelHop++: one workgroup = one (batch, 32x32-input tile).
// Stride-2 2x2 windows => zero halo; all intermediates live in LDS.
__global__ __launch_bounds__(256)
void pixelhop_fused(const float* __restrict__ x,
                    const float* __restrict__ w1,
                    const float* __restrict__ w2,
                    const float* __restrict__ w3,
                    const float* __restrict__ w4,
                    float* __restrict__ o1,
                    float* __restrict__ o2,
                    float* __restrict__ o3,
                    float* __restrict__ o4)
{
  __shared__ float sIn[3][32][32];   // 12 KB input tile
  __shared__ float s1[4][16][16];    //  4 KB hop1 out
  __shared__ float s2[16][8][8];     //  4 KB hop2 out
  __shared__ float s3[64][4][4];     //  4 KB hop3 out

  const int b    = blockIdx.y;       // batch 0..127
  const int ty   = blockIdx.x >> 3;  // tile row 0..7
  const int tx   = blockIdx.x & 7;   // tile col 0..7
  const int tid  = threadIdx.x;
  const int lane = tid & 31;
  const int wave = tid >> 5;         // 8 waves
  const int l15  = lane & 15;        // M (A rows) / N (B,D cols)
  const int kh   = lane >> 4;        // K-half selector (ISA A/B layout)

  // ---- stage input tile [3][32][32] (streamed once -> non-temporal) ----
  {
    const float* xb = x + (size_t)b * 3u * 256u * 256u;
    float* sflat = &sIn[0][0][0];
    for (int i = tid; i < 3 * 32 * 32; i += 256) {
      int c  = i >> 10;
      int yy = (i >> 5) & 31;
      int xx = i & 31;
      sflat[i] = __builtin_nontemporal_load(
          &xb[((size_t)c * 256 + (ty * 32 + yy)) * 256 + (tx * 32 + xx)]);
    }
  }
  __syncthreads();

  // ---- hop1: dense 3->4, o1 tile 16x16. GEMM M=16(4 used) K=12 N=16. ----
  // N-tile j = output row j; pixel n = column l15.
  for (int j = wave; j < 16; j += 8) {           // wave-uniform
    v8f acc = {};
#pragma unroll
    for (int s = 0; s < 3; ++s) {                // K step = input channel
      v2f a;
      if (l15 < 4) {                             // w1[o=l15][c=s][tap]
        const float* wp = w1 + (l15 * 3 + s) * 4 + 2 * kh;
        a.x = wp[0]; a.y = wp[1];
      } else { a.x = 0.f; a.y = 0.f; }
      // tap kk = 2*kh + {0,1} -> (dy,dx) = (kh, {0,1})
      v2f bv = *(const v2f*)&sIn[s][2 * j + kh][2 * l15];
      acc = wmma4(a, bv, acc);
    }
#pragma unroll
    for (int v = 0; v < 8; ++v) {
      int c = v + 8 * kh;                        // D row
      if (c < 4) {
        float val = acc[v];
        s1[c][j][l15] = val;
        __builtin_nontemporal_store(val,
          &o1[(((size_t)b * 4 + c) * 128 + (16 * ty + j)) * 128 + (16 * tx + l15)]);
      }
    }
  }
  __syncthreads();

  // ---- hop2: 4 groups of (1->4) == block-diag GEMM M=16 K=16 N=64. ----
  if (wave < 4) {                                // wave-uniform
    int j  = wave;                               // N-tile: pixels 16j..16j+15
    int y2 = 2 * j + (l15 >> 3);
    int x2 = l15 & 7;
    v8f acc = {};
#pragma unroll
    for (int s = 0; s < 4; ++s) {                // K step = input channel
      v2f a;
      if ((l15 >> 2) == s) {                     // block-diagonal sparsity
        const float* wp = w2 + l15 * 4 + 2 * kh;
        a.x = wp[0]; a.y = wp[1];
      } else { a.x = 0.f; a.y = 0.f; }
      v2f bv = *(const v2f*)&s1[s][2 * y2 + kh][2 * x2];
      acc = wmma4(a, bv, acc);
    }
#pragma unroll
    for (int v = 0; v < 8; ++v) {
      int c = v + 8 * kh;
      float val = acc[v];
      s2[c][y2][x2] = val;
      __builtin_nontemporal_store(val,
        &o2[(((size_t)b * 16 + c) * 64 + (8 * ty + y2)) * 64 + (8 * tx + x2)]);
    }
  }
  __syncthreads();

  // ---- hop3: 16 groups of (1->4) == 4 blocks of M=16 K=16 N=16. ----
  if (wave < 4) {                                // wave-uniform
    int g  = wave;                               // out ch 16g..16g+15, in ch 4g..4g+3
    int y3 = l15 >> 2;
    int x3 = l15 & 3;
    v8f acc = {};
#pragma unroll
    for (int s = 0; s < 4; ++s) {
      v2f a;
      if ((l15 >> 2) == s) {
        const float* wp = w3 + (16 * g + l15) * 4 + 2 * kh;
        a.x = wp[0]; a.y = wp[1];
      } else { a.x = 0.f; a.y = 0.f; }
      v2f bv = *(const v2f*)&s2[4 * g + s][2 * y3 + kh][2 * x3];
      acc = wmma4(a, bv, acc);
    }
#pragma unroll
    for (int v = 0; v < 8; ++v) {
      int cc  = 16 * g + v + 8 * kh;
      float val = acc[v];
      s3[cc][y3][x3] = val;
      __builtin_nontemporal_store(val,
        &o3[(((size_t)b * 64 + cc) * 32 + (4 * ty + y3)) * 32 + (4 * tx + x3)]);
    }
  }
  __syncthreads();

  // ---- hop4: 64 groups of (1->4) == 16 blocks of M=16 K=16 N=4(of 16). ----
  for (int g = wave; g < 16; g += 8) {           // wave-uniform
    int nn = (l15 < 4) ? l15 : 0;                // clamp; cols 4..15 are don't-care
    int y4 = nn >> 1;
    int x4 = nn & 1;
    v8f acc = {};
#pragma unroll
    for (int s = 0; s < 4; ++s) {
      v2f a;
      if ((l15 >> 2) == s) {
        const float* wp = w4 + (16 * g + l15) * 4 + 2 * kh;
        a.x = wp[0]; a.y = wp[1];
      } else { a.x = 0.f; a.y = 0.f; }
      v2f bv = *(const v2f*)&s3[4 * g + s][2 * y4 + kh][2 * x4];
      acc = wmma4(a, bv, acc);
    }
    if (l15 < 4) {
#pragma unroll
      for (int v = 0; v < 8; ++v) {
        int cc = 16 * g + v + 8 * kh;
        __builtin_nontemporal_store(acc[v],
          &o4[(((size_t)b * 256 + cc) * 16 + (2 * ty + y4)) * 16 + (2 * tx + x4)]);
      }
    }
  }
}

extern "C" void kernel_launch(void* const* d_in, const int* in_sizes, int n_in,
                              void* d_out, int out_size, void* d_ws, size_t ws_size,
                              hipStream_t stream) {
  const float* x  = (const float*)d_in[0];   // [128,3,256,256]
  const float* w1 = (const float*)d_in[1];   // [4,3,2,2]
  const float* w2 = (const float*)d_in[2];   // [16,1,2,2]
  const float* w3 = (const float*)d_in[3];   // [64,1,2,2]
  const float* w4 = (const float*)d_in[4];   // [256,1,2,2]

  float* out = (float*)d_out;                // tuple concat, each 8388608 f32
  float* o1 = out;
  float* o2 = out + 8388608;
  float* o3 = out + 16777216;
  float* o4 = out + 25165824;

  dim3 grid(64, 128);                        // 8x8 tiles x 128 batch
  pixelhop_fused<<<grid, 256, 0, stream>>>(x, w1, w2, w3, w4, o1, o2, o3, o4);
}